// Margin_49314814493058
// MI455X (gfx1250) — compile-verified
//
#include <hip/hip_runtime.h>
#include <cstdint>

// ---------------------------------------------------------------------------
// Problem constants (match reference)
// ---------------------------------------------------------------------------
#define BSZ   128
#define DDIM  1024
#define CCLS  1000
#define NOUT  (BSZ * CCLS)       // 128000
#define NHALF (NOUT / 2)         // 64000

typedef __attribute__((ext_vector_type(2))) float v2f;
typedef __attribute__((ext_vector_type(8))) float v8f;
typedef __attribute__((ext_vector_type(4))) int   v4i;

#if __has_builtin(__builtin_amdgcn_wmma_f32_16x16x4_f32)
#define HAVE_WMMA_F32X4 1
#else
#define HAVE_WMMA_F32X4 0
#endif

#if __has_builtin(__builtin_amdgcn_global_load_async_to_lds_b128) && \
    __has_builtin(__builtin_amdgcn_s_wait_asynccnt)
#define HAVE_ASYNC_LDS 1
#else
#define HAVE_ASYNC_LDS 0
#endif

#if HAVE_ASYNC_LDS
// Builtin params are strictly-typed int4 pointers in AS1 (global) / AS3 (LDS).
typedef __attribute__((address_space(1))) v4i gv4i_t;
typedef __attribute__((address_space(3))) v4i lv4i_t;
__device__ __forceinline__ void async_copy_b128(const float* g, float* l) {
  // Generic LDS pointers carry the LDS byte offset in their low 32 bits;
  // generic global pointers equal the 64-bit global address, so integer
  // round-trips produce valid AS1/AS3 pointers for the builtin.
  __builtin_amdgcn_global_load_async_to_lds_b128(
      (gv4i_t*)(uintptr_t)g, (lv4i_t*)(uintptr_t)l, 0, 0);
}
#endif

// ---------------------------------------------------------------------------
// Kernel 1: hash rows.  One wave (32 lanes) per 16 rows.
// sumsq(row) computed as diag(X * X^T) with V_WMMA_F32_16X16X4_F32:
// identical per-lane fragments for A and B are valid because the f32
// 16x16x4 A and B layouts share the (vgpr j, lane-half h) -> K = j + 2h map.
// ---------------------------------------------------------------------------
__global__ __launch_bounds__(32) void hash_rows_wmma(
    const float* __restrict__ src, float* __restrict__ dst, int nrows) {
  __shared__ float ssq[16];
  const int lane = threadIdx.x;     // 0..31 (wave32)
  const int m    = lane & 15;       // matrix row within tile
  const int half = lane >> 4;       // K half select
  const int rowBase = blockIdx.x * 16;
  const int lrow = min(rowBase + m, nrows - 1);   // clamp: keep EXEC all-ones
  const float* rp = src + (size_t)lrow * DDIM + 2 * half;

#if HAVE_WMMA_F32X4
  v8f acc = {0.f, 0.f, 0.f, 0.f, 0.f, 0.f, 0.f, 0.f};
  for (int k0 = 0; k0 < DDIM; k0 += 4) {
    v2f a = *(const v2f*)(rp + k0);   // X[row][k0+2h], X[row][k0+2h+1]
    acc = __builtin_amdgcn_wmma_f32_16x16x4_f32(
        false, a, false, a, (short)0, acc, false, false);
  }
  // D layout: lanes 0-15 -> N=lane, M=r; lanes 16-31 -> N=lane-16, M=r+8.
  // Diagonal M==N==m lives in: lanes 0..7 at acc[m], lanes 24..31 at acc[m-8].
  {
    int want = m - 8 * half;
    float diag = 0.f;
    switch (want) {
      case 0: diag = acc[0]; break;  case 1: diag = acc[1]; break;
      case 2: diag = acc[2]; break;  case 3: diag = acc[3]; break;
      case 4: diag = acc[4]; break;  case 5: diag = acc[5]; break;
      case 6: diag = acc[6]; break;  case 7: diag = acc[7]; break;
      default: break;
    }
    if (want >= 0 && want < 8) ssq[m] = diag;
  }
#else
  // Fallback: butterfly-reduced VALU sumsq (wave32 shuffles).
  for (int r = 0; r < 16; ++r) {
    const float* p = src + (size_t)min(rowBase + r, nrows - 1) * DDIM;
    float s = 0.f;
    for (int c = lane; c < DDIM; c += 32) { float v = p[c]; s += v * v; }
    for (int off = 16; off > 0; off >>= 1) s += __shfl_xor(s, off, 32);
    if (lane == 0) ssq[r] = s;
  }
#endif
  __syncthreads();

  // relu(50 * l2norm) clipped to [0,1]  ==  clamp(x * scale, 0, 1)
  for (int i = lane; i < 16 * DDIM; i += 32) {
    int r = i >> 10;
    int row = rowBase + r;
    if (row >= nrows) break;        // rows ascend with i
    int col = i & (DDIM - 1);
    float sc = 50.f * rsqrtf(fmaxf(ssq[r], 1e-12f));
    float h  = fminf(fmaxf(src[(size_t)row * DDIM + col] * sc, 0.f), 1.f);
    dst[(size_t)row * DDIM + col] = h;
  }
}

// ---------------------------------------------------------------------------
// JAX threefry2x32 (key = (0, seed)), 20 rounds, matches jax.random exactly.
// ---------------------------------------------------------------------------
__device__ __forceinline__ unsigned rotl32(unsigned x, int r) {
  return (x << r) | (x >> (32 - r));
}
__device__ __forceinline__ void threefry2x32(unsigned k0, unsigned k1,
                                             unsigned& x0, unsigned& x1) {
  const unsigned ks2 = 0x1BD11BDAu ^ k0 ^ k1;
  x0 += k0; x1 += k1;
#define TF_R(r) { x0 += x1; x1 = rotl32(x1, r); x1 ^= x0; }
  TF_R(13) TF_R(15) TF_R(26) TF_R(6)   x0 += k1;  x1 += ks2 + 1u;
  TF_R(17) TF_R(29) TF_R(16) TF_R(24)  x0 += ks2; x1 += k0  + 2u;
  TF_R(13) TF_R(15) TF_R(26) TF_R(6)   x0 += k0;  x1 += k1  + 3u;
  TF_R(17) TF_R(29) TF_R(16) TF_R(24)  x0 += k1;  x1 += ks2 + 4u;
  TF_R(13) TF_R(15) TF_R(26) TF_R(6)   x0 += ks2; x1 += k0  + 5u;
#undef TF_R
}

// XLA float32 erf_inv (Giles polynomial, same coefficients as XLA's ErfInv32)
__device__ __forceinline__ float erfinv_f32(float x) {
  float w = -logf((1.0f - x) * (1.0f + x));
  float p;
  if (w < 5.0f) {
    w -= 2.5f;
    p =              2.81022636e-08f;
    p = fmaf(p, w,   3.43273939e-07f);
    p = fmaf(p, w,  -3.5233877e-06f);
    p = fmaf(p, w,  -4.39150654e-06f);
    p = fmaf(p, w,   0.00021858087f);
    p = fmaf(p, w,  -0.00125372503f);
    p = fmaf(p, w,  -0.00417768164f);
    p = fmaf(p, w,   0.246640727f);
    p = fmaf(p, w,   1.50140941f);
  } else {
    w = sqrtf(w) - 3.0f;
    p =             -0.000200214257f;
    p = fmaf(p, w,   0.000100950558f);
    p = fmaf(p, w,   0.00134934322f);
    p = fmaf(p, w,  -0.00367342844f);
    p = fmaf(p, w,   0.00573950773f);
    p = fmaf(p, w,  -0.0076224613f);
    p = fmaf(p, w,   0.00943887047f);
    p = fmaf(p, w,   1.00167406f);
    p = fmaf(p, w,   2.83297682f);
  }
  return p * x;
}

// ---------------------------------------------------------------------------
// Kernel 2: L1 distance + margin + scale.
// 256 threads per block = one 16(B) x 16(C) output tile; D staged in 64-wide
// chunks through LDS via gfx1250 async global->LDS b128 copies.
// ---------------------------------------------------------------------------
#define DCHUNK 64
#define LSTRIDE 68   // 64 + 4 pad floats: kills the 16-way bank conflict

__global__ __launch_bounds__(256) void l1_margin_logits(
    const float* __restrict__ xh, const float* __restrict__ wh,
    const int* __restrict__ labels, float* __restrict__ out) {
  __shared__ float xs[16][LSTRIDE];
  __shared__ float ws[16][LSTRIDE];

  const int t  = threadIdx.x;
  const int tx = t & 15;             // c within tile
  const int ty = t >> 4;             // b within tile
  const int b0 = blockIdx.y * 16;
  const int c0 = blockIdx.x * 16;

  // staging assignment: thread t moves one float4 of each tile per chunk
  const int srow = t >> 4;           // 0..15
  const int sch  = t & 15;           // 0..15 float4 chunks of 64 floats
  const int xrow = b0 + srow;                       // always < 128
  const int wrow = min(c0 + srow, CCLS - 1);        // clamp edge tile

  float acc = 0.f;
  for (int d0 = 0; d0 < DDIM; d0 += DCHUNK) {
    const float* gx = xh + (size_t)xrow * DDIM + d0 + sch * 4;
    const float* gw = wh + (size_t)wrow * DDIM + d0 + sch * 4;
    float* lx = &xs[srow][sch * 4];
    float* lw = &ws[srow][sch * 4];
#if HAVE_ASYNC_LDS
    async_copy_b128(gx, lx);
    async_copy_b128(gw, lw);
    __builtin_amdgcn_s_wait_asynccnt(0);
#else
    *(float4*)lx = *(const float4*)gx;
    *(float4*)lw = *(const float4*)gw;
#endif
    if (d0 + DCHUNK < DDIM) {        // warm L2/L0 for the next chunk
      __builtin_prefetch(gx + DCHUNK, 0, 0);
      __builtin_prefetch(gw + DCHUNK, 0, 0);
    }
    __syncthreads();
#pragma unroll
    for (int d = 0; d < DCHUNK; d += 4) {
      float4 xv = *(const float4*)&xs[ty][d];
      float4 wv = *(const float4*)&ws[tx][d];
      acc += fabsf(xv.x - wv.x) + fabsf(xv.y - wv.y) +
             fabsf(xv.z - wv.z) + fabsf(xv.w - wv.w);
    }
    __syncthreads();
  }

  const int b = b0 + ty;
  const int c = c0 + tx;
  if (c < CCLS) {
    const int idx = b * CCLS + c;
    // jax.random.normal(key(1213)) bit-exact path: threefry counter split
    const bool lohalf = idx < NHALF;
    unsigned x0 = lohalf ? (unsigned)idx : (unsigned)(idx - NHALF);
    unsigned x1 = x0 + (unsigned)NHALF;
    threefry2x32(0u, 1213u, x0, x1);
    unsigned bits = lohalf ? x0 : x1;
    float u01 = __uint_as_float((bits >> 9) | 0x3f800000u) - 1.0f;  // [0,1)
    const float lo = -0.99999994f;                  // nextafter(-1, 0)
    float u  = fmaxf(lo, u01 * (1.0f - lo) + lo);
    float mr = 0.5f + 0.05f * 1.41421356237309515f * erfinv_f32(u);
    float dist  = 48.0f - acc;
    float logit = (labels[b] == c) ? dist : (dist + mr);
    out[idx] = logit * 32.0f;
  }
}

// ---------------------------------------------------------------------------
extern "C" void kernel_launch(void* const* d_in, const int* in_sizes, int n_in,
                              void* d_out, int out_size, void* d_ws,
                              size_t ws_size, hipStream_t stream) {
  (void)in_sizes; (void)n_in; (void)out_size; (void)ws_size;
  const float* feature = (const float*)d_in[0];   // [128, 1024] f32
  const float* W       = (const float*)d_in[1];   // [1000, 1024] f32
  const int*   labels  = (const int*)d_in[2];     // [128] i32
  float* out = (float*)d_out;                     // [128, 1000] f32

  float* xh = (float*)d_ws;                 // 128*1024 f32 = 512 KB
  float* wh = xh + (size_t)BSZ * DDIM;      // 1000*1024 f32 = 4 MB

  hash_rows_wmma<<<dim3((BSZ  + 15) / 16), dim3(32), 0, stream>>>(feature, xh, BSZ);
  hash_rows_wmma<<<dim3((CCLS + 15) / 16), dim3(32), 0, stream>>>(W, wh, CCLS);
  l1_margin_logits<<<dim3((CCLS + 15) / 16, BSZ / 16), dim3(256), 0, stream>>>(
      xh, wh, labels, out);
}